// Loss_8778913153414
// MI455X (gfx1250) — compile-verified
//
#include <hip/hip_runtime.h>
#include <math.h>

#define B_ 32
#define M_ 1024
#define N_ 1024
#define KF 20   // padded feature length (17 used)
#define NK 5    // 5 chained K=4 WMMAs -> K=20
#define MT 4    // m-tiles (of 16 rows) per wave: B fragments reused 4x

typedef __attribute__((ext_vector_type(2))) float v2f;
typedef __attribute__((ext_vector_type(8))) float v8f;

// ---- quaternion (xyzw) -> row-major rotation matrix, matching reference ----
__device__ __forceinline__ void quat_to_R(const float* q4, float R[3][3]) {
    float qx = q4[0], qy = q4[1], qz = q4[2], qw = q4[3];
    float nrm = __builtin_amdgcn_sqrtf(qx*qx + qy*qy + qz*qz + qw*qw) + 1e-8f;
    float inv = 1.f / nrm;
    qx *= inv; qy *= inv; qz *= inv; qw *= inv;
    R[0][0] = 1.f - 2.f*(qy*qy + qz*qz); R[0][1] = 2.f*(qx*qy - qz*qw);       R[0][2] = 2.f*(qx*qz + qy*qw);
    R[1][0] = 2.f*(qx*qy + qz*qw);       R[1][1] = 1.f - 2.f*(qx*qx + qz*qz); R[1][2] = 2.f*(qy*qz - qx*qw);
    R[2][0] = 2.f*(qx*qz - qy*qw);       R[2][1] = 2.f*(qy*qz + qx*qw);       R[2][2] = 1.f - 2.f*(qx*qx + qy*qy);
}

// ---- per-(b,m) feature vector A (length KF) ----
// d2[m,n] = dot(A_m, B_n):
//   A[0..8]  = -2*R[i][j]          B[0..8]  = model[i]*target[j]
//   A[9..11] =  2*(R t)[i]         B[9..11] = model[i]
//   A[12..14]= -2*t[i]             B[12..14]= target[i]
//   A[15]    = |t|^2               B[15]    = 1
//   A[16]    = 1                   B[16]    = |model|^2 + |target|^2
__global__ void featA_kernel(const float* __restrict__ pred_r,
                             const float* __restrict__ pred_t,
                             const float* __restrict__ points,
                             float* __restrict__ featA) {
    int id = blockIdx.x * blockDim.x + threadIdx.x;   // b*M + m
    if (id >= B_ * M_) return;
    float R[3][3];
    quat_to_R(pred_r + (size_t)id * 4, R);
    float t[3];
    #pragma unroll
    for (int i = 0; i < 3; i++) t[i] = points[(size_t)id*3 + i] + pred_t[(size_t)id*3 + i];
    float* A = featA + (size_t)id * KF;
    #pragma unroll
    for (int i = 0; i < 3; i++)
        #pragma unroll
        for (int j = 0; j < 3; j++)
            A[3*i + j] = -2.f * R[i][j];
    #pragma unroll
    for (int i = 0; i < 3; i++) {
        float u = R[i][0]*t[0] + R[i][1]*t[1] + R[i][2]*t[2];
        A[9 + i]  =  2.f * u;
        A[12 + i] = -2.f * t[i];
    }
    A[15] = t[0]*t[0] + t[1]*t[1] + t[2]*t[2];
    A[16] = 1.f;
    A[17] = 0.f; A[18] = 0.f; A[19] = 0.f;
}

__global__ void featB_kernel(const float* __restrict__ model_points,
                             const float* __restrict__ target,
                             float* __restrict__ featB) {
    int id = blockIdx.x * blockDim.x + threadIdx.x;   // b*N + n
    if (id >= B_ * N_) return;
    float mp[3], tg[3];
    #pragma unroll
    for (int i = 0; i < 3; i++) { mp[i] = model_points[(size_t)id*3 + i]; tg[i] = target[(size_t)id*3 + i]; }
    float* Bv = featB + (size_t)id * KF;
    #pragma unroll
    for (int i = 0; i < 3; i++)
        #pragma unroll
        for (int j = 0; j < 3; j++)
            Bv[3*i + j] = mp[i] * tg[j];
    #pragma unroll
    for (int i = 0; i < 3; i++) { Bv[9 + i] = mp[i]; Bv[12 + i] = tg[i]; }
    Bv[15] = 1.f;
    Bv[16] = mp[0]*mp[0] + mp[1]*mp[1] + mp[2]*mp[2] + tg[0]*tg[0] + tg[1]*tg[1] + tg[2]*tg[2];
    Bv[17] = 0.f; Bv[18] = 0.f; Bv[19] = 0.f;
}

// ---- main WMMA kernel ----
// One wave handles MT=4 consecutive 16-row m-tiles; for each 16-col n-tile the
// B fragments are loaded once and fed to 4 chains of 5 V_WMMA_F32_16X16X4_F32.
// sqrt(|d2|): the abs is a free source modifier on V_SQRT_F32 (d2 >= 0 up to
// rounding noise, so |.| == clamp at the precision that matters).  All 8 sqrts
// issued into independent temps so the TRANS hazard slots fill with real work.
__global__ void __launch_bounds__(32) dis_kernel(const float* __restrict__ featA,
                                                 const float* __restrict__ featB,
                                                 float* __restrict__ dis_h) {
    int b    = blockIdx.y;
    int m0   = blockIdx.x * (16 * MT);
    int lane = threadIdx.x;
    int row  = lane & 15;
    int hi   = lane >> 4;

    // A fragments for MT m-tiles (16x4 f32 layout: lanes 0-15 K=4k+{0,1}, lanes 16-31 K=4k+{2,3})
    v2f a[MT][NK];
    #pragma unroll
    for (int mt = 0; mt < MT; mt++) {
        const float* Ab = featA + (size_t)(b * M_ + m0 + mt * 16 + row) * KF + 2 * hi;
        #pragma unroll
        for (int k = 0; k < NK; k++) { a[mt][k][0] = Ab[4*k]; a[mt][k][1] = Ab[4*k + 1]; }
    }

    float acc[MT][8];
    #pragma unroll
    for (int mt = 0; mt < MT; mt++)
        #pragma unroll
        for (int r = 0; r < 8; r++) acc[mt][r] = 0.f;

    const float* Bcol = featB + (size_t)(b * N_ + row) * KF + 2 * hi;
    for (int tile = 0; tile < N_ / 16; ++tile) {
        const float* Bb = Bcol + (size_t)tile * 16 * KF;
        if (tile + 1 < N_ / 16)
            __builtin_prefetch(Bb + (size_t)16 * KF, 0, 3);   // prefetch next n-tile into near cache
        v2f bv[NK];
        #pragma unroll
        for (int k = 0; k < NK; k++) { bv[k][0] = Bb[4*k]; bv[k][1] = Bb[4*k + 1]; }

        #pragma unroll
        for (int mt = 0; mt < MT; mt++) {
            v8f c = {};
            #pragma unroll
            for (int k = 0; k < NK; k++) {
                c = __builtin_amdgcn_wmma_f32_16x16x4_f32(
                        /*neg_a=*/false, a[mt][k], /*neg_b=*/false, bv[k],
                        /*c_mod=*/(short)0, c, /*reuse_a=*/false, /*reuse_b=*/false);
            }
            float sq[8];
            #pragma unroll
            for (int r = 0; r < 8; r++)
                sq[r] = __builtin_amdgcn_sqrtf(__builtin_fabsf(c[r]));  // v_sqrt_f32 v,|v|
            #pragma unroll
            for (int r = 0; r < 8; r++)
                acc[mt][r] += sq[r];
        }
    }

    // C/D layout: VGPR r holds row (tile_m + r) in lanes 0-15 and (tile_m + 8 + r)
    // in lanes 16-31, column = lane&15.  Sum the 16 columns of each half.
    #pragma unroll
    for (int mt = 0; mt < MT; mt++) {
        #pragma unroll
        for (int r = 0; r < 8; r++) {
            float v = acc[mt][r];
            v += __shfl_xor(v, 1, 16);
            v += __shfl_xor(v, 2, 16);
            v += __shfl_xor(v, 4, 16);
            v += __shfl_xor(v, 8, 16);
            if (row == 0)
                dis_h[(size_t)b * M_ + m0 + mt * 16 + 8 * hi + r] = v * (1.f / (float)N_);
        }
    }
}

// ---- per-batch loss partial sum + first-occurrence argmax of clamped pred_c ----
__global__ void reduce_kernel(const float* __restrict__ pred_c,
                              const float* __restrict__ dis_h,
                              const float* __restrict__ w_ptr,
                              float* __restrict__ bsum,
                              float* __restrict__ bdis,
                              int*   __restrict__ bwm) {
    __shared__ float s_sum[256];
    __shared__ float s_max[256];
    __shared__ int   s_idx[256];
    int b = blockIdx.x, tid = threadIdx.x;
    float w = w_ptr[0];
    const float LN2 = 0.69314718055994530942f;
    float sum = 0.f, cmax = -1.f; int cidx = 0;
    for (int m = tid; m < M_; m += 256) {
        float c = fmaxf(pred_c[(size_t)b * M_ + m], 1e-6f);
        float d = dis_h[(size_t)b * M_ + m];
        sum += d * c - w * (LN2 * __builtin_amdgcn_logf(c));   // ln(c) = ln2*log2(c)
        if (c > cmax) { cmax = c; cidx = m; }
    }
    s_sum[tid] = sum; s_max[tid] = cmax; s_idx[tid] = cidx;
    __syncthreads();
    for (int s = 128; s > 0; s >>= 1) {
        if (tid < s) {
            s_sum[tid] += s_sum[tid + s];
            if (s_max[tid + s] > s_max[tid] ||
                (s_max[tid + s] == s_max[tid] && s_idx[tid + s] < s_idx[tid])) {
                s_max[tid] = s_max[tid + s]; s_idx[tid] = s_idx[tid + s];
            }
        }
        __syncthreads();
    }
    if (tid == 0) {
        bsum[b] = s_sum[0];
        bwm[b]  = s_idx[0];
        bdis[b] = dis_h[(size_t)b * M_ + s_idx[0]];
    }
}

// ---- single-wave final reduction: loss and dis_best ----
__global__ void __launch_bounds__(32) final_kernel(const float* __restrict__ bsum,
                                                   const float* __restrict__ bdis,
                                                   float* __restrict__ out) {
    int lane = threadIdx.x;  // 0..31 == batch index
    float s = bsum[lane];
    float d = bdis[lane];
    #pragma unroll
    for (int off = 16; off >= 1; off >>= 1) {
        s += __shfl_xor(s, off, 32);
        d += __shfl_xor(d, off, 32);
    }
    if (lane == 0) {
        out[0] = s / (float)(B_ * M_);
        out[1] = d / (float)B_;
    }
}

// ---- best-pose transform: new_points / new_target ----
__global__ void transform_kernel(const float* __restrict__ pred_r,
                                 const float* __restrict__ pred_t,
                                 const float* __restrict__ points,
                                 const float* __restrict__ target,
                                 const int*   __restrict__ bwm,
                                 float* __restrict__ out_np,
                                 float* __restrict__ out_nt) {
    int b   = blockIdx.y;
    int idx = blockIdx.x * blockDim.x + threadIdx.x;  // 0..1023 (M_==N_)
    if (idx >= M_) return;
    int wm = bwm[b];
    float R[3][3];
    quat_to_R(pred_r + (size_t)(b * M_ + wm) * 4, R);
    float t[3];
    #pragma unroll
    for (int i = 0; i < 3; i++)
        t[i] = points[(size_t)(b * M_ + wm) * 3 + i] + pred_t[(size_t)(b * M_ + wm) * 3 + i];

    float p[3], tg[3];
    #pragma unroll
    for (int i = 0; i < 3; i++) {
        p[i]  = points[(size_t)(b * M_ + idx) * 3 + i] - t[i];
        tg[i] = target[(size_t)(b * N_ + idx) * 3 + i] - t[i];
    }
    #pragma unroll
    for (int j = 0; j < 3; j++) {
        out_np[(size_t)(b * M_ + idx) * 3 + j] = p[0]*R[0][j]  + p[1]*R[1][j]  + p[2]*R[2][j];
        out_nt[(size_t)(b * N_ + idx) * 3 + j] = tg[0]*R[0][j] + tg[1]*R[1][j] + tg[2]*R[2][j];
    }
}

extern "C" void kernel_launch(void* const* d_in, const int* in_sizes, int n_in,
                              void* d_out, int out_size, void* d_ws, size_t ws_size,
                              hipStream_t stream) {
    const float* pred_r       = (const float*)d_in[0];
    const float* pred_t       = (const float*)d_in[1];
    const float* pred_c       = (const float*)d_in[2];
    const float* target       = (const float*)d_in[3];
    const float* model_points = (const float*)d_in[4];
    // d_in[5] = idx (unused in refine path), d_in[8] = refine (unused)
    const float* points       = (const float*)d_in[6];
    const float* w            = (const float*)d_in[7];
    float* out = (float*)d_out;

    float* featA = (float*)d_ws;                              // B*M*KF
    float* featB = featA + (size_t)B_ * M_ * KF;              // B*N*KF
    float* dis_h = featB + (size_t)B_ * N_ * KF;              // B*M
    float* bsum  = dis_h + (size_t)B_ * M_;                   // B
    float* bdis  = bsum + B_;                                 // B
    int*   bwm   = (int*)(bdis + B_);                         // B

    featA_kernel<<<(B_ * M_ + 255) / 256, 256, 0, stream>>>(pred_r, pred_t, points, featA);
    featB_kernel<<<(B_ * N_ + 255) / 256, 256, 0, stream>>>(model_points, target, featB);
    dis_kernel<<<dim3(M_ / (16 * MT), B_), 32, 0, stream>>>(featA, featB, dis_h);
    reduce_kernel<<<B_, 256, 0, stream>>>(pred_c, dis_h, w, bsum, bdis, bwm);
    final_kernel<<<1, 32, 0, stream>>>(bsum, bdis, out);
    transform_kernel<<<dim3(M_ / 256, B_), 256, 0, stream>>>(
        pred_r, pred_t, points, target, bwm,
        out + 2, out + 2 + (size_t)B_ * M_ * 3);
}